// LearnableBinsQuantizer_7541962572331
// MI455X (gfx1250) — compile-verified
//
#include <hip/hip_runtime.h>
#include <hip/hip_bf16.h>

// LearnableBinsQuantizer forward: out[i] = bins[rintf(clamp(x[i],0,1)*255)]
// 512 MiB in + 512 MiB out of fp32 -> pure HBM-bound streaming gather
// (~46 us floor at 23.3 TB/s). No matmul semantics -> WMMA inapplicable.
// CDNA5 paths used: wave32 blocks, b128 non-temporal vmem, LDS-staged bin
// table with per-lane ds_load gathers, global_prefetch_b8.

typedef float f4 __attribute__((ext_vector_type(4)));

#define NBINS 256
#define BLOCK 256

__global__ __launch_bounds__(BLOCK) void
lbq_kernel(const float* __restrict__ x,
           const float* __restrict__ bins,
           float* __restrict__ out,
           int n4, int n) {
    __shared__ float lds_bins[NBINS];
    // blockDim.x == NBINS == 256: one table element per thread.
    lds_bins[threadIdx.x] = bins[threadIdx.x];
    __syncthreads();

    const f4* __restrict__ xv = (const f4*)x;
    f4* __restrict__ ov = (f4*)out;

    // 32-bit indexing: n <= 2^31 (in_sizes is int), so n4 <= 2^29 and all
    // byte offsets fit comfortably -> no 64-bit adds/compares in the loop.
    const int stride = (int)(gridDim.x * BLOCK);
    int i = (int)(blockIdx.x * BLOCK + threadIdx.x);

    for (; i < n4; i += stride) {
        // Speculative prefetch of this thread's next chunk; out-of-range
        // translations are silently dropped (ISA 10.5) -> no branch needed.
        __builtin_prefetch((const void*)&xv[i + stride], 0, 0);

        // Streamed once -> non-temporal (TH_NT) 128-bit load.
        f4 v = __builtin_nontemporal_load(&xv[i]);
        f4 r;
#pragma unroll
        for (int k = 0; k < 4; ++k) {
            float xc = __builtin_fminf(__builtin_fmaxf(v[k], 0.0f), 1.0f);
            // round-half-to-even, matches jnp.round -> v_rndne_f32
            int idx = (int)__builtin_rintf(xc * 255.0f);   // in [0,255] by construction
            r[k] = lds_bins[idx];                           // ds_load_b32 gather
        }
        // Streamed once -> non-temporal (TH_NT) 128-bit store.
        __builtin_nontemporal_store(r, &ov[i]);
    }

    // Scalar tail for n % 4 != 0 (not hit for this shape, kept for safety).
    int ti = n4 * 4 + (int)(blockIdx.x * BLOCK + threadIdx.x);
    if (ti < n) {
        float xc = __builtin_fminf(__builtin_fmaxf(x[ti], 0.0f), 1.0f);
        int idx = (int)__builtin_rintf(xc * 255.0f);
        out[ti] = lds_bins[idx];
    }
}

extern "C" void kernel_launch(void* const* d_in, const int* in_sizes, int n_in,
                              void* d_out, int out_size, void* d_ws, size_t ws_size,
                              hipStream_t stream) {
    (void)n_in; (void)d_ws; (void)ws_size; (void)out_size;
    const float* x    = (const float*)d_in[0];   // (16, 8192, 1024) fp32
    const float* bins = (const float*)d_in[1];   // (256,) fp32
    float* out        = (float*)d_out;           // same shape as x, fp32

    const int n  = in_sizes[0];
    const int n4 = n / 4;

    // ~8 float4 iterations per thread at full size: amortizes the LDS table
    // fill and gives every WGP many wave32 waves for latency hiding.
    int want = (n4 + BLOCK - 1) / BLOCK;
    int blocks = want < 1 ? 1 : (want > 16384 ? 16384 : want);

    lbq_kernel<<<blocks, BLOCK, 0, stream>>>(x, bins, out, n4, n);
}